// DKWinners_14362370638087
// MI455X (gfx1250) — compile-verified
//
#include <hip/hip_runtime.h>
#include <math.h>

// ---------------------------------------------------------------------------
// DKWinners (duty-cycle boosted k-winner mask), MI455X / gfx1250.
//
//   boost[n]          = exp((1/16 - duty[n]) * bs)
//   window w (0..4095): argmax_j  x[b, 15w+j] * boost[15w+j],  j in [0,16)
//   out[b, 16w+j]     = (j == argmax) * x[b, 16w+j]   (all other j -> 0)
//
// Memory-bound: ~256 MB HBM traffic -> ~11 us floor @ 23.3 TB/s; compute is
// kept off the critical path:
//   * CDNA5 async global->LDS copies (global_load_async_to_lds_b128,
//     ASYNCcnt) stage the 4-byte-aligned stride-15 window region with fully
//     aligned 128B bursts (tile byte base = 15360*wtile).
//   * stride-15 LDS reads are bank-conflict-free (gcd(15,64)=1).
//   * exp((td-dc)*bs) = exp2((td-dc)*bs*log2e): one v_fma + one v_exp_f32
//     (TRANS, co-executes with VALU) per element instead of libm expf.
//   * output = NT b128 zero-fill + single b32 winner patch (same-wave stores
//     complete in order, ISA 7.3/10.1); NT keeps the write-once output from
//     evicting x (128 MB) out of the 192 MB L2 so x is read from HBM ~once.
// ---------------------------------------------------------------------------

#define OUT_DIM       4096
#define DPC           16
#define NCOL          (OUT_DIM * DPC)   // 65536
#define BATCH         512
#define WPB           256               // windows per block (= threads)
#define TILE_FLOATS   4096              // floats staged per array per block

typedef float v4f __attribute__((ext_vector_type(4)));   // native vector: NT-store OK

__global__ __launch_bounds__(256)
void dkwinners_kernel(const float* __restrict__ x,
                      const float* __restrict__ duty,
                      const float* __restrict__ bstr,
                      float* __restrict__ out)
{
    __shared__ __align__(16) float sx[TILE_FLOATS];   // x window region tile
    __shared__ __align__(16) float sd[TILE_FLOATS];   // duty_cycle tile

    const int t     = threadIdx.x;          // 0..255
    const int wtile = blockIdx.x;           // 0..15
    const int b     = blockIdx.y;           // 0..511
    const int W0    = wtile * WPB;          // first window of this tile
    const int gwin  = (DPC - 1) * W0;       // = 15*W0, window-region element offset

    const float* xrow = x + (size_t)b * NCOL;

    // ---- Stage window-region x and duty into LDS (async, ASYNCcnt) --------
    // Tile spans [gwin, gwin+4096); max gwin = 15*3840 = 57600, and
    // 57600+4096 = 61696 <= 65536, so always in bounds for x-row and duty.
    {
        const unsigned long long xbase = (unsigned long long)(const void*)xrow;
        const unsigned long long dbase = (unsigned long long)(const void*)duty;
        #pragma unroll
        for (int k = 0; k < 4; ++k) {
            const int      v4   = t + k * 256;                      // float4 idx 0..1023
            const unsigned voff = (unsigned)((gwin + v4 * 4) * 4);  // byte offset
            const unsigned lx   = (unsigned)(size_t)(&sx[v4 * 4]);  // LDS byte addr
            const unsigned ld   = (unsigned)(size_t)(&sd[v4 * 4]);
            asm volatile("global_load_async_to_lds_b128 %0, %1, %2"
                         :: "v"(lx), "v"(voff), "s"(xbase) : "memory");
            asm volatile("global_load_async_to_lds_b128 %0, %1, %2"
                         :: "v"(ld), "v"(voff), "s"(dbase) : "memory");
        }
    }
    asm volatile("s_wait_asynccnt 0" ::: "memory");  // my async copies landed
    __syncthreads();                                  // everyone's landed

    // ---- Per-window boosted argmax (first-max tie-break, like jnp.argmax) --
    // boost = exp2((td - dc) * s), s = bs*log2(e); arg = fma(-dc, s, td*s).
    const float s  = bstr[0] * 1.4426950408889634f;       // uniform (SGPR) scalar
    const float c  = ((float)OUT_DIM / (float)NCOL) * s;  // td*s
    const int   l0 = (DPC - 1) * t;                       // 15*t, conflict-free

    float best = -INFINITY;
    int   amax = 0;
    #pragma unroll
    for (int j = 0; j < DPC; ++j) {
        const float e = __builtin_fmaf(-sd[l0 + j], s, c);  // v_fma_f32
        const float p = __builtin_amdgcn_exp2f(e);          // v_exp_f32 (TRANS)
        const float v = sx[l0 + j] * p;
        if (v > best) { best = v; amax = j; }   // strict '>' keeps first max
    }

    // ---- Output: zero-fill window, then patch the single winner ------------
    // 16*w*4 bytes is 64B aligned: NT b128 zero stores; then one b32 load of
    // the winner x and one b32 store. Same-wave stores retire in order, so
    // the patch lands after the zero-fill.
    const int    w   = W0 + t;
    const size_t go  = (size_t)DPC * (size_t)w;
    float*       orow = out + (size_t)b * NCOL;
    v4f*         oo4  = (v4f*)(orow + go);

    const v4f z = {0.0f, 0.0f, 0.0f, 0.0f};
    #pragma unroll
    for (int q = 0; q < 4; ++q)
        __builtin_nontemporal_store(z, &oo4[q]);            // NT b128 stream

    const float xwin = xrow[go + (size_t)amax];             // single winner read
    __builtin_nontemporal_store(xwin, &orow[go + (size_t)amax]);
}

extern "C" void kernel_launch(void* const* d_in, const int* in_sizes, int n_in,
                              void* d_out, int out_size, void* d_ws, size_t ws_size,
                              hipStream_t stream) {
    (void)in_sizes; (void)n_in; (void)out_size; (void)d_ws; (void)ws_size;
    const float* x    = (const float*)d_in[0];   // (512, 65536) f32
    const float* duty = (const float*)d_in[1];   // (65536,)     f32
    const float* bstr = (const float*)d_in[2];   // (1,)         f32
    float*       out  = (float*)d_out;           // (512, 65536) f32

    dim3 grid(OUT_DIM / WPB, BATCH);             // (16, 512)
    dkwinners_kernel<<<grid, 256, 0, stream>>>(x, duty, bstr, out);
}